// TemperedCrossEntropyLoss_39006892982398
// MI455X (gfx1250) — compile-verified
//
#include <hip/hip_runtime.h>
#include <hip/hip_bf16.h>
#include <cstdint>

// Tempered cross-entropy loss, N=4096 rows x V=32000 vocab, fp32.
// Strategy (MI455X): memory-bound problem -> single HBM pass.
// One workgroup per row; stage the 128KB row into LDS (320KB/WGP available)
// with CDNA5 async global->LDS loads, then run the 5 logical passes
// (max, 3x za, final sum) entirely out of LDS.

#define NT    1024          // threads per block (32 wave32 waves)
#define VOCAB 32000
#define VEC4  (VOCAB / 4)   // 8000 float4 per row

typedef int v4i __attribute__((ext_vector_type(4)));
typedef __attribute__((address_space(1))) v4i* gv4i_ptr;   // global 16B vector ptr
typedef __attribute__((address_space(3))) v4i* lv4i_ptr;   // LDS    16B vector ptr

// ---------- CDNA5-specific helpers (builtin if present, asm fallback) ----------

__device__ __forceinline__ void async_copy_f4(const float4* gsrc, float4* ldst) {
#if defined(__AMDGCN__) && __has_builtin(__builtin_amdgcn_global_load_async_to_lds_b128)
    __builtin_amdgcn_global_load_async_to_lds_b128(
        (gv4i_ptr)(uintptr_t)gsrc,
        (lv4i_ptr)(uintptr_t)ldst,
        /*imm offset*/0, /*cpol*/0);
#else
    unsigned           lds_addr = (unsigned)(uintptr_t)ldst;  // low 32 bits = LDS addr
    unsigned long long gaddr    = (unsigned long long)(uintptr_t)gsrc;
    asm volatile("global_load_async_to_lds_b128 %0, %1, off"
                 :: "v"(lds_addr), "v"(gaddr) : "memory");
#endif
}

__device__ __forceinline__ void wait_async_zero() {
#if defined(__AMDGCN__) && __has_builtin(__builtin_amdgcn_s_wait_asynccnt)
    __builtin_amdgcn_s_wait_asynccnt(0);
#else
    asm volatile("s_wait_asynccnt 0" ::: "memory");
#endif
}

__device__ __forceinline__ float fast_exp2(float x) {   // v_exp_f32: 2^x
#if defined(__AMDGCN__) && __has_builtin(__builtin_amdgcn_exp2f)
    return __builtin_amdgcn_exp2f(x);
#else
    return exp2f(x);
#endif
}

__device__ __forceinline__ float fast_log2(float x) {   // v_log_f32: log2(x)
#if defined(__AMDGCN__) && __has_builtin(__builtin_amdgcn_logf)
    return __builtin_amdgcn_logf(x);
#else
    return log2f(x);
#endif
}

// ---------- block reduction (wave32 shuffle + LDS, 32 waves) ----------

__device__ __forceinline__ float blk_reduce(float v, bool do_max) {
    __shared__ float red[33];
    const int lane = threadIdx.x & 31;
    const int wid  = threadIdx.x >> 5;
#pragma unroll
    for (int o = 16; o > 0; o >>= 1) {
        float other = __shfl_xor(v, o, 32);
        v = do_max ? fmaxf(v, other) : (v + other);
    }
    __syncthreads();                 // protect red[] from previous use
    if (lane == 0) red[wid] = v;
    __syncthreads();
    if (wid == 0) {
        float t = red[lane];         // blockDim = 1024 -> exactly 32 waves
#pragma unroll
        for (int o = 16; o > 0; o >>= 1) {
            float other = __shfl_xor(t, o, 32);
            t = do_max ? fmaxf(t, other) : (t + other);
        }
        if (lane == 0) red[32] = t;
    }
    __syncthreads();
    return red[32];
}

// ---------- kernel ----------

__global__ __launch_bounds__(NT) void tempered_ce_kernel(
    const float* __restrict__ x, const int* __restrict__ y, float* __restrict__ out)
{
    __shared__ float4 sx4[VEC4];     // 128 000 B row cache in LDS

    const int row = blockIdx.x;
    const int tid = threadIdx.x;
    const float4* xrow4 = (const float4*)(x + (size_t)row * VOCAB);

    // Stage the full row into LDS with async DMA-style loads (ASYNCcnt).
    for (int i = tid; i < VEC4; i += NT)
        async_copy_f4(xrow4 + i, &sx4[i]);
    wait_async_zero();
    __syncthreads();

    // Constants: t1=0.7, t2=1.3
    const float c2   = 1.0f - 1.3f;            // -0.3
    const float e_za = 1.0f / c2;              // -10/3 : exp_t exponent
    const float e_s  = e_za * (2.0f - 0.7f);   // -13/3 : p^{2-t1} exponent

    // Pass 1: row max (mu)
    float m = -INFINITY;
    for (int i = tid; i < VEC4; i += NT) {
        float4 v = sx4[i];
        m = fmaxf(m, fmaxf(fmaxf(v.x, v.y), fmaxf(v.z, v.w)));
    }
    const float mu = blk_reduce(m, true);

    // Passes 2-4: za fixed-point iterations.
    // base = 1 + (1-t2)*w*(x-mu) = fma(c2*w, x, 1 - c2*w*mu); base >= 1 always.
    float w = 1.0f, za = 1.0f;
#pragma unroll
    for (int it = 0; it < 3; ++it) {
        const float k  = c2 * w;
        const float b0 = 1.0f - k * mu;
        float acc = 0.0f;
        for (int i = tid; i < VEC4; i += NT) {
            float4 v = sx4[i];
            acc += fast_exp2(e_za * fast_log2(fmaxf(fmaf(k, v.x, b0), 1e-12f)));
            acc += fast_exp2(e_za * fast_log2(fmaxf(fmaf(k, v.y, b0), 1e-12f)));
            acc += fast_exp2(e_za * fast_log2(fmaxf(fmaf(k, v.z, b0), 1e-12f)));
            acc += fast_exp2(e_za * fast_log2(fmaxf(fmaf(k, v.w, b0), 1e-12f)));
        }
        za = blk_reduce(acc, false);
        w  = fast_exp2(c2 * fast_log2(za));    // za^{1-t2} = za^{-0.3}
    }

    // lambda = (za^{0.3} - 1)/0.3 + mu
    const float lambda = (fast_exp2(-c2 * fast_log2(za)) - 1.0f) / (-c2) + mu;

    // Pass 5: s = sum_j p_j^{2-t1} = sum_j base_j^{-13/3},  base = fma(c2, x, 1 - c2*lambda)
    const float b2 = 1.0f - c2 * lambda;
    float acc = 0.0f;
    for (int i = tid; i < VEC4; i += NT) {
        float4 v = sx4[i];
        acc += fast_exp2(e_s * fast_log2(fmaxf(fmaf(c2, v.x, b2), 1e-12f)));
        acc += fast_exp2(e_s * fast_log2(fmaxf(fmaf(c2, v.y, b2), 1e-12f)));
        acc += fast_exp2(e_s * fast_log2(fmaxf(fmaf(c2, v.z, b2), 1e-12f)));
        acc += fast_exp2(e_s * fast_log2(fmaxf(fmaf(c2, v.w, b2), 1e-12f)));
    }
    const float s = blk_reduce(acc, false);

    if (tid == 0) {
        const float* sx = (const float*)sx4;
        const int yi = y[row];
        const float base_y = fmaxf(fmaf(c2, sx[yi], b2), 1e-12f);
        // p_y^{1-t1} = base_y^{(1-t1)/(1-t2)} = base_y^{-1}
        const float term1 = (1.0f - 1.0f / base_y) / (1.0f - 0.7f);  // -log_t(p_y, t1)
        out[row] = term1 - (1.0f - s) / (2.0f - 0.7f);
    }
}

// ---------- launcher ----------

extern "C" void kernel_launch(void* const* d_in, const int* in_sizes, int n_in,
                              void* d_out, int out_size, void* d_ws, size_t ws_size,
                              hipStream_t stream) {
    const float* x = (const float*)d_in[0];
    const int*   y = (const int*)d_in[1];
    float*       o = (float*)d_out;
    const int N = out_size;   // one loss per row (4096)
    (void)in_sizes; (void)n_in; (void)d_ws; (void)ws_size;
    tempered_ce_kernel<<<dim3(N), dim3(NT), 0, stream>>>(x, y, o);
}